// EarlyRewardLoss_26388279067028
// MI455X (gfx1250) — compile-verified
//
#include <hip/hip_runtime.h>

// EarlyRewardLoss for MI455X (gfx1250, wave32).
// Strategy: memory-bound gather+scan+reduce.
//   Kernel 1: one wave32 per row n. Lanes scan T=365 timesteps in chunks of 32:
//     - wave-level prefix-product scan (Hillis-Steele via shfl_up) for the
//       cumprod(1-d) "budget", carrying a running product across chunks,
//     - fused gather logp[n,t,y] + contribution, per-wave reduction -> ws[n].
//   Kernel 2: exact deterministic 4096->1 reduction using
//     V_WMMA_F32_16X16X4_F32 with B = all-ones (layout-independent),
//     A = 64 per-lane f32 partials in the documented 16x4 A layout.
//     Loads are b128 + unrolled so multiple loads are in flight (the previous
//     version serialized one b32 load + s_wait_loadcnt 0 per element).

#define ALPHA_F   0.5f
#define EPSILON_F 10.0f

typedef __attribute__((ext_vector_type(2))) float v2f;
typedef __attribute__((ext_vector_type(8))) float v8f;

__global__ __launch_bounds__(256) void erl_row_kernel(
    const float* __restrict__ logp,     // (N,T,C)
    const float* __restrict__ deltas,   // (N,T)
    const int*   __restrict__ y_true,   // (N,T)
    float*       __restrict__ row_out,  // (N,)
    int N, int T, int C)
{
  const int lane = threadIdx.x & 31;
  const int n    = blockIdx.x * (blockDim.x >> 5) + (threadIdx.x >> 5);
  if (n >= N) return;  // whole wave exits together (same n across lanes)

  const float* drow = deltas + (size_t)n * T;
  const int*   yrow = y_true + (size_t)n * T;
  const float* lrow = logp   + (size_t)n * T * C;
  const float  eps_t = EPSILON_F / (float)T;
  const float  inv_T = 1.0f / (float)T;

  float running = 1.0f;   // product of q over all previous chunks
  float acc     = 0.0f;

  for (int base = 0; base < T; base += 32) {
    const int t = base + lane;

    // q[t] = 1 - deltas[t+1] for t < T-1 ; q[T-1] = 0 (so d=1 captures the
    // final comp[:, -1] term) ; q = 1 beyond T (identity for the scan).
    float q;
    if (t < T - 1)       q = 1.0f - drow[t + 1];
    else if (t == T - 1) q = 0.0f;
    else                 q = 1.0f;

    // Inclusive prefix product within the wave.
    float incl = q;
    #pragma unroll
    for (int off = 1; off < 32; off <<= 1) {
      float other = __shfl_up(incl, off, 32);
      if (lane >= off) incl *= other;
    }
    // Exclusive prefix product.
    float excl = __shfl_up(incl, 1, 32);
    if (lane == 0) excl = 1.0f;

    if (t < T) {
      // Pt = d[t] * prod_{j<t} q[j] + eps/T   (d[t] = 1 - q[t])
      const float pt  = (1.0f - q) * (running * excl) + eps_t;
      const int   cls = yrow[t];
      const float lp  = lrow[(size_t)t * C + cls];
      const float w   = 1.0f - (float)t * inv_T;
      // ALPHA*(-lp)*Pt  -  (1-ALPHA)*Pt*exp(lp)*w
      acc += pt * (ALPHA_F * (-lp) - (1.0f - ALPHA_F) * __expf(lp) * w);
    }
    running *= __shfl(incl, 31, 32);  // fold chunk product into carry
  }

  // Deterministic wave reduction.
  #pragma unroll
  for (int off = 16; off >= 1; off >>= 1) acc += __shfl_xor(acc, off, 32);
  if (lane == 0) row_out[n] = acc;
}

__global__ __launch_bounds__(32) void erl_reduce_wmma_kernel(
    const float* __restrict__ row_sums, float* __restrict__ out, int N)
{
  const int lane = threadIdx.x & 31;

  // Two deterministic per-lane partials, fed by b128 loads.
  // Any fixed assignment of elements to the 64 (lane, a0/a1) slots yields the
  // same exact total through the WMMA reduction below.
  float a0 = 0.0f, a1 = 0.0f;

  const int N4 = N >> 2;                       // # of float4 chunks
  const float4* rs4 = (const float4*)row_sums; // row_sums is 16B-aligned (d_ws)

  // Unroll x4: up to 4 independent b128 loads in flight per lane before use.
  #pragma unroll 4
  for (int i = lane; i < N4; i += 32) {
    const float4 v = rs4[i];
    a0 += v.x + v.y;
    a1 += v.z + v.w;
  }
  // Tail (N not a multiple of 4): deterministic, lane 0 only.
  if (lane == 0) {
    for (int i = N4 << 2; i < N; ++i) a0 += row_sums[i];
  }

  // A (16x4 f32, documented layout): lane L<16 -> A[L][0],A[L][1];
  // lane L>=16 -> A[L-16][2],A[L-16][3].  B = all ones -> D[m][*] = rowsum_m.
  v2f A; A[0] = a0; A[1] = a1;
  v2f B; B[0] = 1.0f; B[1] = 1.0f;
  v8f Cm = {};
  v8f D = __builtin_amdgcn_wmma_f32_16x16x4_f32(
      /*neg_a=*/false, A, /*neg_b=*/false, B,
      /*c_mod=*/(short)0, Cm, /*reuse_a=*/false, /*reuse_b=*/false);

  // Per C/D layout: lanes 0-15 hold rowsum[0..7] in D[0..7],
  // lanes 16-31 hold rowsum[8..15]. Sum registers, then fold halves.
  float s = D[0] + D[1] + D[2] + D[3] + D[4] + D[5] + D[6] + D[7];
  s += __shfl_xor(s, 16, 32);

  if (lane == 0) out[0] = s / (float)N;  // mean over rows
}

extern "C" void kernel_launch(void* const* d_in, const int* in_sizes, int n_in,
                              void* d_out, int out_size, void* d_ws, size_t ws_size,
                              hipStream_t stream) {
  const float* logp   = (const float*)d_in[0];  // (N,T,C) f32
  const float* deltas = (const float*)d_in[1];  // (N,T)   f32
  const int*   y_true = (const int*)  d_in[2];  // (N,T)   i32
  float*       out    = (float*)d_out;

  const int NTC = in_sizes[0];
  const int NT  = in_sizes[1];
  const int C   = NTC / NT;   // 32
  const int T   = 365;        // fixed by reference
  const int N   = NT / T;     // 4096

  float* row_sums = (float*)d_ws;  // N floats of scratch

  const int waves_per_block = 8;   // 256 threads = 8 wave32
  const int blocks = (N + waves_per_block - 1) / waves_per_block;
  erl_row_kernel<<<blocks, 256, 0, stream>>>(logp, deltas, y_true, row_sums, N, T, C);
  erl_reduce_wmma_kernel<<<1, 32, 0, stream>>>(row_sums, out, N);
}